// IGLMDALayer_30064771072071
// MI455X (gfx1250) — compile-verified
//
#include <hip/hip_runtime.h>
#include <hip/hip_bf16.h>

// s[i,j] = mish(u[i] + v[j]),  u = h_m@(w_m@a1[:128]) + m_h@a1[128:],  etc.
// N1=4096 rows (u / i), N2=8192 rows (v / j), F=256, H=128, OUT=128.

typedef __attribute__((ext_vector_type(2))) float v2f;
typedef __attribute__((ext_vector_type(8))) float v8f;

#define N1 4096
#define N2 8192
#define FD 256
#define HH 128

// ---------------------------------------------------------------------------
// Kernel 1: wa_m[k] = sum_j w_m[k][j]*a[j]   (k<256, j<128)
//           wa_d[k] = sum_j w_d[k][j]*a[256+j]
// ---------------------------------------------------------------------------
__global__ __launch_bounds__(512) void wa_kernel(const float* __restrict__ wm,
                                                 const float* __restrict__ wd,
                                                 const float* __restrict__ a,
                                                 float* __restrict__ ws) {
  int t = threadIdx.x;                 // 0..511
  const float* w = (t < 256) ? wm : wd;
  const float* av = a + ((t < 256) ? 0 : 256);
  int k = t & 255;
  float s = 0.0f;
#pragma unroll 8
  for (int j = 0; j < 128; ++j) s = fmaf(w[k * 128 + j], av[j], s);
  ws[t] = s;                           // ws[0..255]=wa_m, ws[256..511]=wa_d
}

// ---------------------------------------------------------------------------
// Kernel 2: one wave32 computes 16 rows of u (or v) with V_WMMA_F32_16X16X4.
// A[i,k] = blended row data, B[k,n] = wa[k] broadcast over n -> every column
// of D holds the same dot product. A/B K striping per ISA: lane L holds
// K = (L>>4)*2 + {0,1} in VGPRs {0,1}.
// ---------------------------------------------------------------------------
__global__ __launch_bounds__(32) void uv_kernel(
    const float* __restrict__ m,  const float* __restrict__ m1,
    const float* __restrict__ m2, const float* __restrict__ mh,
    const float* __restrict__ d,  const float* __restrict__ d1,
    const float* __restrict__ d2, const float* __restrict__ dh,
    const float* __restrict__ am, const float* __restrict__ ad,
    const float* __restrict__ wa_ws, const float* __restrict__ avec,
    float* __restrict__ uv) {
  int bid = blockIdx.x;                // 0..255 -> u, 256..767 -> v
  int lane = threadIdx.x;
  bool isV = bid >= (N1 / 16);
  int rowbase = (isV ? (bid - N1 / 16) : bid) * 16;

  const float* s0 = isV ? d : m;
  const float* s1 = isV ? d1 : m1;
  const float* s2 = isV ? d2 : m2;
  const float* sh = isV ? dh : mh;
  const float* ac = isV ? ad : am;
  const float* wa = wa_ws + (isV ? 256 : 0);
  const float* ah = avec + (isV ? 384 : 128);   // a1[128:256] / a2[128:256]

  float c0 = ac[0], c1 = ac[1], c2 = ac[2];

  int half = lane >> 4;                // 0 or 1 (selects K pair)
  int r = lane & 15;                   // row within tile
  int rowoff = (rowbase + r) * FD;
  int rowoffh = (rowbase + r) * HH;

  v8f acc = {};

  // K = 0..255 over blended matrix vs wa
#pragma unroll 4
  for (int kb = 0; kb < FD; kb += 4) {
    int k0 = kb + half * 2;
    v2f av, bv;
    av.x = fmaf(c0, s0[rowoff + k0],
            fmaf(c1, s1[rowoff + k0], c2 * s2[rowoff + k0]));
    av.y = fmaf(c0, s0[rowoff + k0 + 1],
            fmaf(c1, s1[rowoff + k0 + 1], c2 * s2[rowoff + k0 + 1]));
    bv.x = wa[k0];
    bv.y = wa[k0 + 1];
    acc = __builtin_amdgcn_wmma_f32_16x16x4_f32(false, av, false, bv,
                                                (short)0, acc, false, false);
  }
  // K = 0..127 over head matrix vs a[128:]/a[384:]
#pragma unroll 4
  for (int kb = 0; kb < HH; kb += 4) {
    int k0 = kb + half * 2;
    v2f av, bv;
    av.x = sh[rowoffh + k0];
    av.y = sh[rowoffh + k0 + 1];
    bv.x = ah[k0];
    bv.y = ah[k0 + 1];
    acc = __builtin_amdgcn_wmma_f32_16x16x4_f32(false, av, false, bv,
                                                (short)0, acc, false, false);
  }

  // D layout: VGPR g, lanes 0-15: M=g; lanes 16-31: M=8+g (all N equal).
  if (r == 0) {
    float* out = uv + (isV ? N1 : 0) + rowbase + 8 * half;
    out[0] = acc[0]; out[1] = acc[1]; out[2] = acc[2]; out[3] = acc[3];
    out[4] = acc[4]; out[5] = acc[5]; out[6] = acc[6]; out[7] = acc[7];
  }
}

// ---------------------------------------------------------------------------
// Kernel 3: s[i,j] = mish(u[i]+v[j]).  Pure streaming-store kernel.
// mish(x) = x * (t^2+2t)/(t^2+2t+2), t = exp(min(x,20))  (exact identity;
// clamp exact in f32 since tanh(20) rounds to 1).
// ---------------------------------------------------------------------------
__device__ __forceinline__ float mish1(float x) {
  float xc = fminf(x, 20.0f);
  float t = __expf(xc);                          // v_exp_f32 path
  float n = fmaf(t, t, t + t);                   // t^2 + 2t
  float den = n + 2.0f;
  float r0 = __builtin_amdgcn_rcpf(den);         // v_rcp_f32
  float rr = r0 * fmaf(-den, r0, 2.0f);          // one Newton step
  return x * (n * rr);
}

#define RROWS 16
__global__ __launch_bounds__(256) void mish_outer_kernel(
    const float* __restrict__ uv, float* __restrict__ out) {
  const float* u = uv;
  const float* v = uv + N1;
  int j = (blockIdx.x * 256 + threadIdx.x) * 4;     // 0..8191 step 4
  int i0 = blockIdx.y * RROWS;

  float4 vv = *(const float4*)(v + j);
  float* o = out + (size_t)i0 * N2 + j;
#pragma unroll
  for (int rr = 0; rr < RROWS; ++rr) {
    float ui = u[i0 + rr];
    float4 res;
    res.x = mish1(ui + vv.x);
    res.y = mish1(ui + vv.y);
    res.z = mish1(ui + vv.z);
    res.w = mish1(ui + vv.w);
    *(float4*)o = res;                              // global_store_b128
    o += N2;
  }
}

// ---------------------------------------------------------------------------
extern "C" void kernel_launch(void* const* d_in, const int* in_sizes, int n_in,
                              void* d_out, int out_size, void* d_ws,
                              size_t ws_size, hipStream_t stream) {
  const float* m   = (const float*)d_in[0];
  const float* m1  = (const float*)d_in[1];
  const float* m2  = (const float*)d_in[2];
  const float* mh  = (const float*)d_in[3];
  const float* d   = (const float*)d_in[4];
  const float* d1  = (const float*)d_in[5];
  const float* d2  = (const float*)d_in[6];
  const float* dh  = (const float*)d_in[7];
  const float* wm  = (const float*)d_in[8];
  const float* wd  = (const float*)d_in[9];
  const float* am  = (const float*)d_in[10];
  const float* ad  = (const float*)d_in[11];
  const float* a   = (const float*)d_in[12];

  float* ws = (float*)d_ws;       // [0,512): wa_m|wa_d, [512,12800): u|v
  float* uv = ws + 512;
  float* out = (float*)d_out;

  wa_kernel<<<1, 512, 0, stream>>>(wm, wd, a, ws);
  uv_kernel<<<(N1 + N2) / 16, 32, 0, stream>>>(m, m1, m2, mh, d, d1, d2, dh,
                                               am, ad, ws, a, uv);
  mish_outer_kernel<<<dim3(N2 / (256 * 4), N1 / RROWS), 256, 0, stream>>>(uv,
                                                                          out);
}